// HWT_48876727828663
// MI455X (gfx1250) — compile-verified
//
#include <hip/hip_runtime.h>

// Gray+Haar fusion for MI455X (gfx1250, wave32).
// Memory-bound: 201 MB in + 67 MB out -> ~11.5us floor at 23.3 TB/s.
// Data path: GLOBAL_LOAD_ASYNC_TO_LDS_B128 (ASYNCcnt) -> s_wait_asynccnt ->
// ds_load_b128 -> VALU butterfly -> non-temporal b64 stores.

typedef float f32x4 __attribute__((ext_vector_type(4)));
typedef float f32x2 __attribute__((ext_vector_type(2)));
typedef int v4i __attribute__((vector_size(16)));  // matches builtin param type

#define AS_LDS __attribute__((address_space(3)))
#define AS_GLOBAL __attribute__((address_space(1)))

#if __has_builtin(__builtin_amdgcn_global_load_async_to_lds_b128)
#define HWT_HAVE_ASYNC_LDS 1
#else
#define HWT_HAVE_ASYNC_LDS 0
#endif

__global__ __launch_bounds__(256) void hwt_gray_haar_kernel(
    const float* __restrict__ x, float* __restrict__ out) {
  const int tid = threadIdx.x;            // 0..255: two output columns each
  const int b   = blockIdx.x >> 9;        // batch image
  const int r   = blockIdx.x & 511;       // output row
  const size_t plane = 1024u * 1024u;     // input channel plane (elements)

  // Input base for this thread: 4 consecutive pixels (16B aligned) of row 2r.
  const float* base =
      x + (size_t)b * 3u * plane + (size_t)(2 * r) * 1024u + 4u * (unsigned)tid;

  f32x4 v[6];  // [c0r0, c0r1, c1r0, c1r1, c2r0, c2r1]

#if HWT_HAVE_ASYNC_LDS
  __shared__ f32x4 tile[6 * 256];  // 24KB; layout (k*256+tid) -> conflict-free b128
  AS_LDS f32x4* lt = (AS_LDS f32x4*)tile;
#pragma unroll
  for (int c = 0; c < 3; ++c) {
#pragma unroll
    for (int rr = 0; rr < 2; ++rr) {
      const int k = c * 2 + rr;
      const float* g = base + (size_t)c * plane + (size_t)rr * 1024u;
      // async B128: memory -> LDS, tracked with ASYNCcnt (no VGPR return path)
      __builtin_amdgcn_global_load_async_to_lds_b128(
          (AS_GLOBAL v4i*)g, (AS_LDS v4i*)&lt[k * 256 + tid], /*offset=*/0,
          /*cpol=*/0);
    }
  }
#if __has_builtin(__builtin_amdgcn_s_wait_asynccnt)
  __builtin_amdgcn_s_wait_asynccnt(0);
#else
  asm volatile("s_wait_asynccnt 0" ::: "memory");
#endif
  // Each lane reads back only its own staged data: no barrier required.
#pragma unroll
  for (int k = 0; k < 6; ++k) v[k] = tile[k * 256 + tid];
#else
  // Fallback: direct non-temporal B128 loads (read-once stream).
#pragma unroll
  for (int c = 0; c < 3; ++c) {
#pragma unroll
    for (int rr = 0; rr < 2; ++rr) {
      v[c * 2 + rr] = __builtin_nontemporal_load(
          (const f32x4*)(base + (size_t)c * plane + (size_t)rr * 1024u));
    }
  }
#endif

  // BGR weights with the Haar 0.5 folded in.
  const float w0 = 0.114f * 0.5f;
  const float w1 = 0.587f * 0.5f;
  const float w2 = 0.299f * 0.5f;

  float gt[4], gb[4];  // gray of top / bottom input rows, 4 pixels each
#pragma unroll
  for (int i = 0; i < 4; ++i) {
    gt[i] = fmaf(w2, v[4][i], fmaf(w1, v[2][i], w0 * v[0][i]));
    gb[i] = fmaf(w2, v[5][i], fmaf(w1, v[3][i], w0 * v[1][i]));
  }

  // Haar butterfly for the two 2x2 blocks this thread owns.
  const float s0 = gt[0] + gt[1], s1 = gb[0] + gb[1];
  const float d0 = gt[0] - gt[1], d1 = gb[0] - gb[1];
  const float t0 = gt[2] + gt[3], t1 = gb[2] + gb[3];
  const float e0 = gt[2] - gt[3], e1 = gb[2] - gb[3];

  f32x2 cA = {s0 + s1, t0 + t1};
  f32x2 cH = {s0 - s1, t0 - t1};
  f32x2 cV = {d0 + d1, e0 + e1};
  f32x2 cD = {d0 - d1, e0 - e1};

  // out layout (16, 4, 512, 512); plane stride 512*512.
  float* o = out + ((size_t)b * 2048u + (size_t)r) * 512u + 2u * (unsigned)tid;
  __builtin_nontemporal_store(cA, (f32x2*)(o));
  __builtin_nontemporal_store(cH, (f32x2*)(o + 262144));
  __builtin_nontemporal_store(cV, (f32x2*)(o + 524288));
  __builtin_nontemporal_store(cD, (f32x2*)(o + 786432));
}

extern "C" void kernel_launch(void* const* d_in, const int* in_sizes, int n_in,
                              void* d_out, int out_size, void* d_ws,
                              size_t ws_size, hipStream_t stream) {
  (void)in_sizes; (void)n_in; (void)d_ws; (void)ws_size; (void)out_size;
  const float* x = (const float*)d_in[0];
  float* out = (float*)d_out;
  // One block per (batch, output row): 16 * 512 = 8192 blocks of 256 threads.
  hipLaunchKernelGGL(hwt_gray_haar_kernel, dim3(16 * 512), dim3(256), 0, stream,
                     x, out);
}